// SecondOrderNetwork_13984413516481
// MI455X (gfx1250) — compile-verified
//
#include <hip/hip_runtime.h>

typedef __attribute__((ext_vector_type(2))) float v2f;
typedef __attribute__((ext_vector_type(8))) float v8f;

#define BSZ   65536
#define CDIM  512
#define ALPHAF 0.05f

// ---- GEMM tiling ----
#define BM   32     // M rows per workgroup
#define BN   256    // N cols per workgroup (2 blocks cover C=512)
#define KT   16     // K slice staged in LDS per iteration
#define LDA  18     // lsA row stride (floats): even (b64-aligned), conflict-free
#define LDB  20     // lsB row stride (floats): 80B rows -> float4-aligned, conflict-free

// Fused: comparison_out = rate*relu(A@Wc^T + bc) + (1-rate)*prev
//        wager += comparison_out @ Ww^T   (atomic partial sums)
__global__ __launch_bounds__(256)
void son_gemm_kernel(const float* __restrict__ A,      // comparison_matrix [B,C]
                     const float* __restrict__ prev,   // prev_comparison  [B,C]
                     const float* __restrict__ rate_p, // cascade_rate [1]
                     const float* __restrict__ Wc,     // W_comp [C,C]
                     const float* __restrict__ bc,     // b_comp [C]
                     const float* __restrict__ Ww,     // W_wager [2,C]
                     float* __restrict__ wager,        // [B,2] pre-init with b_wager
                     float* __restrict__ comp_out)     // [B,C]
{
    __shared__ float lsA[BM * LDA];   // A tile: lsA[m*LDA + k]
    __shared__ float lsB[BN * LDB];   // W^T tile, n-major: lsB[n*LDB + k] = Wc[n0+n][k0+k]

    const int nBlock = blockIdx.x & 1;
    const int mBlock = blockIdx.x >> 1;
    const int m0 = mBlock * BM;
    const int n0 = nBlock * BN;

    const int tid   = threadIdx.x;
    const int lane  = tid & 31;
    const int wave  = tid >> 5;
    const int waveM = wave & 1;    // 2 M-subtiles of 16
    const int waveN = wave >> 1;   // 4 N-strips of 64

    const float rate = rate_p[0];

    v8f acc[4];
    #pragma unroll
    for (int i = 0; i < 4; ++i) { v8f z = {}; acc[i] = z; }

    // staging assignments
    const int aRow = tid >> 3;            // 0..31
    const int aK   = (tid & 7) << 1;      // 0,2,..,14
    const float* Arow = A  + (size_t)(m0 + aRow) * CDIM;
    const float* Wrow = Wc + (size_t)(n0 + tid)  * CDIM;

    // WMMA fragment lane mapping (ISA 7.12.2, 32-bit A 16x4 / B 4x16)
    const int koff = (lane >> 4) << 1;            // half 0 -> K{0,1}, half 1 -> K{2,3}
    const int mloc = waveM * 16 + (lane & 15);    // A-frag: M = lane%16
    const int nlan = lane & 15;                   // B/C/D: N = lane%16

    for (int k0 = 0; k0 < CDIM; k0 += KT) {
        if (k0 + KT < CDIM)
            __builtin_prefetch(Arow + k0 + KT + aK, 0, 1);  // global_prefetch_b8

        __syncthreads();
        // stage A tile: 32 rows x 16 K, 8B per thread
        *(v2f*)&lsA[aRow * LDA + aK] = *(const v2f*)(Arow + k0 + aK);
        // stage B tile: one W_comp row per thread, kept row-contiguous (n-major)
        #pragma unroll
        for (int q = 0; q < 4; ++q) {
            *(float4*)&lsB[tid * LDB + q * 4] = *(const float4*)(Wrow + k0 + q * 4);
        }
        __syncthreads();

        #pragma unroll
        for (int kk = 0; kk < KT; kk += 4) {
            // A-frag: contiguous ds_load_b64 (stride-18 rows: conflict-free)
            v2f a = *(const v2f*)&lsA[mloc * LDA + kk + koff];
            #pragma unroll
            for (int nt = 0; nt < 4; ++nt) {
                const int nb = waveN * 64 + nt * 16 + nlan;
                // B-frag: contiguous ds_load_b64 (stride-20 rows: conflict-free)
                v2f b = *(const v2f*)&lsB[nb * LDB + kk + koff];
                acc[nt] = __builtin_amdgcn_wmma_f32_16x16x4_f32(
                    false, a, false, b, (short)0, acc[nt], false, false);
            }
        }
    }

    // ---- epilogue: bias + relu + cascade blend + store + wager partials ----
    const int half = lane >> 4;   // C/D: VGPR v -> M = v + 8*half, N = lane%16

    float ws0[8], ws1[8];
    #pragma unroll
    for (int v = 0; v < 8; ++v) { ws0[v] = 0.f; ws1[v] = 0.f; }

    #pragma unroll
    for (int nt = 0; nt < 4; ++nt) {
        const int gn  = n0 + waveN * 64 + nt * 16 + nlan;
        const float bcn = bc[gn];
        const float w0n = Ww[gn];
        const float w1n = Ww[CDIM + gn];
        #pragma unroll
        for (int v = 0; v < 8; ++v) {
            const int gm = m0 + waveM * 16 + v + half * 8;
            float h = acc[nt][v] + bcn;
            h = h > 0.f ? h : 0.f;
            float o = rate * h + (1.f - rate) * prev[(size_t)gm * CDIM + gn];
            comp_out[(size_t)gm * CDIM + gn] = o;
            ws0[v] += o * w0n;
            ws1[v] += o * w1n;
        }
    }

    #pragma unroll
    for (int v = 0; v < 8; ++v) {
        float s0 = ws0[v], s1 = ws1[v];
        #pragma unroll
        for (int m = 1; m < 16; m <<= 1) {   // reduce over the 16 N-lanes per half
            s0 += __shfl_xor(s0, m, 32);
            s1 += __shfl_xor(s1, m, 32);
        }
        if (nlan == 0) {
            const int gm = m0 + waveM * 16 + v + half * 8;
            atomicAdd(&wager[gm * 2 + 0], s0);
            atomicAdd(&wager[gm * 2 + 1], s1);
        }
    }
}

// wager[i] = b_wager[i%2]  (runs before GEMM atomics; same stream => ordered)
__global__ void son_wager_init(const float* __restrict__ bw, float* __restrict__ wager)
{
    int i = blockIdx.x * blockDim.x + threadIdx.x;
    if (i < BSZ * 2) wager[i] = bw[i & 1];
}

// Sequential EMA over 65536 rewards as a parallel scan of affine maps.
#define TGT_THREADS 1024
#define TGT_CHUNK   64
__global__ __launch_bounds__(TGT_THREADS)
void son_target_kernel(const float* __restrict__ rewards, float* __restrict__ target)
{
    __shared__ float sP[TGT_THREADS];
    __shared__ float sS[TGT_THREADS];
    const int tid  = threadIdx.x;
    const int base = tid * TGT_CHUNK;
    const float a = ALPHAF, oma = 1.f - ALPHAF;

    // local pass from ema=0: chunk acts as x -> P*x + S
    float S = 0.f, P = 1.f;
    for (int e = 0; e < TGT_CHUNK; ++e) {
        S = a * rewards[base + e] + oma * S;
        P *= oma;
    }
    sP[tid] = P; sS[tid] = S;
    __syncthreads();

    // Hillis-Steele inclusive scan; combine(left,right): (Pr*Pl, Pr*Sl + Sr)
    for (int off = 1; off < TGT_THREADS; off <<= 1) {
        float pl = 0.f, sl = 0.f;
        const bool act = (tid >= off);
        if (act) { pl = sP[tid - off]; sl = sS[tid - off]; }
        __syncthreads();
        if (act) { sS[tid] = sP[tid] * sl + sS[tid]; sP[tid] = sP[tid] * pl; }
        __syncthreads();
    }

    float ema = (tid == 0) ? 0.f : sS[tid - 1];  // ema at chunk start
    for (int e = 0; e < TGT_CHUNK; ++e) {
        float g = rewards[base + e];
        ema = a * g + oma * ema;
        const bool gt = g > ema;
        target[(base + e) * 2 + 0] = gt ? 1.f : 0.f;
        target[(base + e) * 2 + 1] = gt ? 0.f : 1.f;
    }
}

extern "C" void kernel_launch(void* const* d_in, const int* in_sizes, int n_in,
                              void* d_out, int out_size, void* d_ws, size_t ws_size,
                              hipStream_t stream)
{
    const float* cmpm = (const float*)d_in[0];  // comparison_matrix [B,C]
    const float* prev = (const float*)d_in[1];  // prev_comparison  [B,C]
    const float* rate = (const float*)d_in[2];  // cascade_rate [1]
    const float* rew  = (const float*)d_in[3];  // rewards [B]
    const float* Wc   = (const float*)d_in[4];  // W_comp [C,C]
    const float* bc   = (const float*)d_in[5];  // b_comp [C]
    const float* Ww   = (const float*)d_in[6];  // W_wager [2,C]
    const float* bw   = (const float*)d_in[7];  // b_wager [2]

    float* out    = (float*)d_out;
    float* wager  = out;                                  // [B,2]
    float* comp   = out + (size_t)BSZ * 2;                // [B,C]
    float* target = comp + (size_t)BSZ * CDIM;            // [B,2]

    son_wager_init<<<(BSZ * 2 + 255) / 256, 256, 0, stream>>>(bw, wager);

    dim3 grid((BSZ / BM) * (CDIM / BN));   // 2048 * 2 = 4096 blocks
    son_gemm_kernel<<<grid, 256, 0, stream>>>(cmpm, prev, rate, Wc, bc, Ww, wager, comp);

    son_target_kernel<<<1, TGT_THREADS, 0, stream>>>(rew, target);
}